// HMMLayer_19464791786228
// MI455X (gfx1250) — compile-verified
//
#include <hip/hip_runtime.h>

// HMM forward-backward posterior for MI455X (gfx1250, wave32, WMMA).
// x:[B,T,N] f32, P:[N,N] f32, p0:[N] f32  ->  out:[B,T,N] f32 posterior.
#define STATES   128
#define BATCH    64
#define TLEN     2048
#define BT       16                 // batch tile = WMMA M
#define NT       16                 // per-wave N tile = WMMA N
#define NWAVES   (STATES / NT)      // 8 waves
#define NTHREADS (NWAVES * 32)      // 256 threads (8 x wave32)
#define KTILES   (STATES / 4)       // 32 chained 16x16x4 WMMAs (K = 128)

typedef __attribute__((ext_vector_type(2))) float v2f;
typedef __attribute__((ext_vector_type(8))) float v8f;

__device__ __forceinline__ float sigmoidf_fast(float v) {
    return 1.0f / (1.0f + __expf(-v));
}

// sum across each 16-lane half of the wave32 (xor masks 1,2,4,8 stay in-half)
__device__ __forceinline__ float halfsum(float v) {
    v += __shfl_xor(v, 1, 32);
    v += __shfl_xor(v, 2, 32);
    v += __shfl_xor(v, 4, 32);
    v += __shfl_xor(v, 8, 32);
    return v;
}

// DIR = 0: forward (writes alpha_t to out)
// DIR = 1: backward (beta scan, fused posterior: out[t] <- normalize(alpha_t * beta_t))
template <int DIR>
__global__ __launch_bounds__(NTHREADS)
void hmm_scan_kernel(const float* __restrict__ x,
                     const float* __restrict__ P,
                     const float* __restrict__ p0,
                     float* __restrict__ out)
{
    __shared__ float lvec[BT * STATES];   // running alpha (fwd) / beta->u (bwd), 8 KB
    __shared__ float rowsum[BT];
    __shared__ float rowsum2[BT];

    const int tid = threadIdx.x;
    const int w   = tid >> 5;            // wave id -> N tile [16w, 16w+16)
    const int l   = tid & 31;
    const int h   = l >> 4;              // half of wave32
    const int n16 = l & 15;
    const int b0  = blockIdx.x * BT;

    // ---- Preload time-invariant B fragments of P (fwd) or P^T (bwd) ----
    // B 4x16 f32 layout: lane l, vgpr v holds B[k = v + 2*(l/16)][n = l%16]
    v2f bfrag[KTILES];
    #pragma unroll
    for (int kt = 0; kt < KTILES; ++kt) {
        #pragma unroll
        for (int v = 0; v < 2; ++v) {
            const int k = kt * 4 + v + 2 * h;
            const int n = NT * w + n16;
            bfrag[kt][v] = (DIR == 0) ? P[k * STATES + n]      // D = A * P
                                      : P[n * STATES + k];     // D = A * P^T
        }
    }

    // C/D 16x16 f32 layout: lane l, vgpr r holds (m = r + 8*(l/16), n = l%16).
    // Flat [b,t,n] offset base (at t = 0) for the 8 rows this lane owns:
    unsigned cst[8];
    #pragma unroll
    for (int r = 0; r < 8; ++r)
        cst[r] = (unsigned)(b0 + r + 8 * h) * (TLEN * STATES) + NT * w + n16;

    if (tid < BT) { rowsum[tid] = 0.0f; rowsum2[tid] = 0.0f; }
    __syncthreads();

    // ---- Initialize the scan state in LDS ----
    if (DIR == 0) {
        // alpha_0 = normalize(p0 * sigmoid(x[:,0,:])); also write to out[:,0,:]
        #pragma unroll
        for (int i = 0; i < (BT * STATES) / NTHREADS; ++i) {
            const int idx = tid + NTHREADS * i;
            const int row = idx >> 7, col = idx & (STATES - 1);
            const float v = p0[col] *
                sigmoidf_fast(x[(unsigned)(b0 + row) * (TLEN * STATES) + col]);
            lvec[idx] = v;
        }
        __syncthreads();
        if (tid < BT) {
            float s = 0.0f;
            for (int c = 0; c < STATES; ++c) s += lvec[tid * STATES + c];
            rowsum[tid] = s;
        }
        __syncthreads();
        #pragma unroll
        for (int i = 0; i < (BT * STATES) / NTHREADS; ++i) {
            const int idx = tid + NTHREADS * i;
            const int row = idx >> 7, col = idx & (STATES - 1);
            const float v = lvec[idx] / rowsum[row];
            lvec[idx] = v;
            out[(unsigned)(b0 + row) * (TLEN * STATES) + col] = v;
        }
        __syncthreads();
        if (tid < BT) rowsum[tid] = 0.0f;
        __syncthreads();
    } else {
        // beta_{T-1} = 1; posterior at T-1 == alpha_{T-1} (already in out)
        #pragma unroll
        for (int i = 0; i < (BT * STATES) / NTHREADS; ++i)
            lvec[tid + NTHREADS * i] = 1.0f;
        __syncthreads();
    }

    // ---- Sequential scan over T (2047 steps) ----
    #pragma unroll 1
    for (int s = 1; s < TLEN; ++s) {
        const int t = (DIR == 0) ? s : (TLEN - s);

        if (DIR == 0) {
            // acc = alpha_{t-1} * P  (8 waves x 32 chained v_wmma_f32_16x16x4_f32)
            v8f acc = {};
            #pragma unroll
            for (int kt = 0; kt < KTILES; ++kt) {
                // A 16x4 f32 layout: lane l, vgpr v = A[l%16][kt*4 + v + 2*(l/16)]
                const v2f a = *reinterpret_cast<const v2f*>(
                    &lvec[n16 * STATES + kt * 4 + 2 * h]);
                acc = __builtin_amdgcn_wmma_f32_16x16x4_f32(
                    false, a, false, bfrag[kt], (short)0, acc, false, false);
            }
            // multiply by y_t = sigmoid(x_t), partial per-row sums
            float vals[8], part[8];
            #pragma unroll
            for (int r = 0; r < 8; ++r) {
                const float y = sigmoidf_fast(x[cst[r] + (unsigned)t * STATES]);
                vals[r] = acc[r] * y;
                part[r] = halfsum(vals[r]);
            }
            __builtin_prefetch(&x[cst[0] + (unsigned)(t + 2) * STATES], 0, 1);
            __syncthreads();                     // alpha reads done; zeros visible
            if (n16 == 0) {
                #pragma unroll
                for (int r = 0; r < 8; ++r)
                    atomicAdd(&rowsum[r + 8 * h], part[r]);
            }
            __syncthreads();                     // rowsum ready
            #pragma unroll
            for (int r = 0; r < 8; ++r) {
                const float v = vals[r] / rowsum[r + 8 * h];
                lvec[(r + 8 * h) * STATES + NT * w + n16] = v;   // alpha_t
                out[cst[r] + (unsigned)t * STATES] = v;
            }
            __syncthreads();                     // writes done, rowsum reads done
            if (tid < BT) rowsum[tid] = 0.0f;
        } else {
            // u = beta_t .* y_t, in place in LDS
            #pragma unroll
            for (int i = 0; i < (BT * STATES) / NTHREADS; ++i) {
                const int idx = tid + NTHREADS * i;
                const int row = idx >> 7, col = idx & (STATES - 1);
                lvec[idx] *= sigmoidf_fast(
                    x[(unsigned)(b0 + row) * (TLEN * STATES) +
                      (unsigned)t * STATES + col]);
            }
            __syncthreads();                     // u ready
            // acc = u * P^T
            v8f acc = {};
            #pragma unroll
            for (int kt = 0; kt < KTILES; ++kt) {
                const v2f a = *reinterpret_cast<const v2f*>(
                    &lvec[n16 * STATES + kt * 4 + 2 * h]);
                acc = __builtin_amdgcn_wmma_f32_16x16x4_f32(
                    false, a, false, bfrag[kt], (short)0, acc, false, false);
            }
            float part[8], alr[8];
            #pragma unroll
            for (int r = 0; r < 8; ++r) {
                part[r] = halfsum(acc[r]);
                alr[r]  = out[cst[r] + (unsigned)(t - 1) * STATES]; // alpha_{t-1}
            }
            if (t >= 2)
                __builtin_prefetch(&x[cst[0] + (unsigned)(t - 2) * STATES], 0, 1);
            __syncthreads();                     // u reads done; zeros visible
            if (n16 == 0) {
                #pragma unroll
                for (int r = 0; r < 8; ++r)
                    atomicAdd(&rowsum[r + 8 * h], part[r]);
            }
            __syncthreads();                     // rowsum ready
            float pa[8];
            #pragma unroll
            for (int r = 0; r < 8; ++r) {
                const float v = acc[r] / rowsum[r + 8 * h];       // beta_{t-1}
                lvec[(r + 8 * h) * STATES + NT * w + n16] = v;
                pa[r] = alr[r] * v;
            }
            #pragma unroll
            for (int r = 0; r < 8; ++r) part[r] = halfsum(pa[r]);
            if (n16 == 0) {
                #pragma unroll
                for (int r = 0; r < 8; ++r)
                    atomicAdd(&rowsum2[r + 8 * h], part[r]);
            }
            __syncthreads();                     // rowsum2 ready, beta stored
            #pragma unroll
            for (int r = 0; r < 8; ++r)
                out[cst[r] + (unsigned)(t - 1) * STATES] =
                    pa[r] / rowsum2[r + 8 * h];                   // posterior_{t-1}
            __syncthreads();                     // rowsum2 reads done
            if (tid < BT) { rowsum[tid] = 0.0f; rowsum2[tid] = 0.0f; }
        }
    }
}

extern "C" void kernel_launch(void* const* d_in, const int* in_sizes, int n_in,
                              void* d_out, int out_size, void* d_ws, size_t ws_size,
                              hipStream_t stream) {
    (void)in_sizes; (void)n_in; (void)out_size; (void)d_ws; (void)ws_size;
    const float* x  = (const float*)d_in[0];
    const float* P  = (const float*)d_in[1];
    const float* p0 = (const float*)d_in[2];
    float* out = (float*)d_out;

    dim3 grid(BATCH / BT);        // 4 independent batch tiles
    dim3 block(NTHREADS);         // 8 x wave32
    hmm_scan_kernel<0><<<grid, block, 0, stream>>>(x, P, p0, out);   // alphas
    hmm_scan_kernel<1><<<grid, block, 0, stream>>>(x, P, p0, out);   // betas + posterior
}